// one_pt_cpu_test_9887014716138
// MI455X (gfx1250) — compile-verified
//
#include <hip/hip_runtime.h>
#include <hip/hip_bf16.h>
#include <math.h>

// ---------------------------------------------------------------------------
// Problem constants (from reference): B=4, D=H=W=64, P=512.
// ---------------------------------------------------------------------------
#define BB     4
#define DHW    (64 * 64 * 64)          // 262144 = 2^18
#define NTOT   (BB * DHW)              // 1,048,576
#define INF_L  (DHW + 1)
#define PP     512
#define NPROP  20                      // fixed propagation launches (graph-safe)
#define NB_SUM 64                      // blocks for the WMMA sum reduction

typedef __attribute__((ext_vector_type(2))) float v2f;
typedef __attribute__((ext_vector_type(4))) float v4f;
typedef __attribute__((ext_vector_type(8))) float v8f;

// ---------------------------------------------------------------------------
// Kernel 1: global sum of logits via V_WMMA_F32_16X16X4_F32 (A=data, B=ones).
// D = A x ones + C  =>  sum(D) == 16 * sum(A); accumulate C across the loop.
// Each lane loads a float4 (global_load_b128) -> 128 elements per wave-iter,
// fed to two WMMAs with independent accumulators (pipelines the XDL chain).
// N is an exact multiple of the sweep (512 waves * 128 = 65536), so the loop
// is branch-free and EXEC stays all-ones at every WMMA, as the ISA requires.
// ---------------------------------------------------------------------------
__global__ void sum_logits_wmma(const float* __restrict__ x, long n,
                                float* __restrict__ partials) {
  v8f c0 = {0.f, 0.f, 0.f, 0.f, 0.f, 0.f, 0.f, 0.f};
  v8f c1 = {0.f, 0.f, 0.f, 0.f, 0.f, 0.f, 0.f, 0.f};
  v2f onesv;
  onesv[0] = 1.0f;
  onesv[1] = 1.0f;

  const int  lane          = threadIdx.x & 31;
  const int  wave          = threadIdx.x >> 5;
  const long wavesPerBlock = blockDim.x >> 5;
  const long gwave         = (long)blockIdx.x * wavesPerBlock + wave;
  const long nwaves        = (long)gridDim.x * wavesPerBlock;

  const v4f* __restrict__ xv = (const v4f*)x;   // 16B-aligned (harness alloc)
  const long stride = nwaves * 128;             // elements per sweep

  for (long base = gwave * 128; base < n; base += stride) {
    const v4f av = xv[(base >> 2) + lane];      // global_load_b128
    v2f a0, a1;
    a0[0] = av[0]; a0[1] = av[1];
    a1[0] = av[2]; a1[1] = av[3];
    // (neg_a, A, neg_b, B, c_mod, C, reuse_a, reuse_b)
    c0 = __builtin_amdgcn_wmma_f32_16x16x4_f32(false, a0, false, onesv,
                                               (short)0, c0, false, false);
    c1 = __builtin_amdgcn_wmma_f32_16x16x4_f32(false, a1, false, onesv,
                                               (short)0, c1, false, false);
  }

  c0 += c1;
  float s = c0[0] + c0[1] + c0[2] + c0[3] + c0[4] + c0[5] + c0[6] + c0[7];
  // wave32 butterfly reduction (fixed order -> deterministic)
  for (int off = 16; off > 0; off >>= 1) s += __shfl_xor(s, off, 32);

  __shared__ float ws[8];
  if (lane == 0) ws[wave] = s;
  __syncthreads();
  if (threadIdx.x == 0) {
    float t = 0.0f;
    for (int w = 0; w < (int)wavesPerBlock; ++w) t += ws[w];
    partials[blockIdx.x] = t * (1.0f / 16.0f);   // undo the x16 from B=ones
  }
}

// ---------------------------------------------------------------------------
// Kernel 2: fold the 64 block partials into the mean; zero per-batch counters.
// ---------------------------------------------------------------------------
__global__ void finalize_mean(const float* __restrict__ partials,
                              float* __restrict__ meanp,
                              int* __restrict__ counts, float inv_n) {
  __shared__ float s[NB_SUM];
  const int t = threadIdx.x;
  s[t] = partials[t];
  __syncthreads();
  for (int o = NB_SUM / 2; o > 0; o >>= 1) {
    if (t < o) s[t] += s[t + o];
    __syncthreads();
  }
  if (t == 0) {
    *meanp = s[0] * inv_n;
    counts[0] = counts[1] = counts[2] = counts[3] = 0;
  }
}

// ---------------------------------------------------------------------------
// Kernel 3: init labels: foreground -> flat-idx-in-batch + 1, background -> INF.
// ---------------------------------------------------------------------------
__global__ void init_labels(const float* __restrict__ logits,
                            const float* __restrict__ meanp,
                            int* __restrict__ L) {
  const int i = blockIdx.x * blockDim.x + threadIdx.x;
  if (i >= NTOT) return;
  const float mean = *meanp;
  const int v = i & (DHW - 1);
  L[i] = (logits[i] > mean) ? (v + 1) : INF_L;
}

// ---------------------------------------------------------------------------
// Kernel 4: one Jacobi iteration of 6-neighbor min + 6 pointer jumps.
// Labels only decrease (monotone); the fixed point is the reference's
// min-flat-index label. In-place updates are safe (any mix of old/new values
// is still a valid same-component label).
// ---------------------------------------------------------------------------
__global__ void propagate(int* __restrict__ L) {
  const int i = blockIdx.x * blockDim.x + threadIdx.x;
  if (i >= NTOT) return;
  int lab = L[i];
  if (lab > DHW) return;                 // background
  const int v = i & (DHW - 1);
  const int base = i - v;                // b * DHW
  const int x = v & 63, y = (v >> 6) & 63, z = v >> 12;

  int m = lab;
  if (x > 0)  m = min(m, L[i - 1]);
  if (x < 63) m = min(m, L[i + 1]);
  if (y > 0)  m = min(m, L[i - 64]);
  if (y < 63) m = min(m, L[i + 64]);
  if (z > 0)  m = min(m, L[i - 4096]);
  if (z < 63) m = min(m, L[i + 4096]);
#pragma unroll
  for (int t = 0; t < 6; ++t)            // pointer jumping: chase toward root
    m = min(m, L[base + m - 1]);
  L[i] = m;
}

// ---------------------------------------------------------------------------
// Kernel 5: count component roots per batch (lab == own idx + 1).
// Each 256-thread block lies entirely inside one batch (DHW % 256 == 0).
// ---------------------------------------------------------------------------
__global__ void count_roots(const int* __restrict__ L, int* __restrict__ counts) {
  const int i = blockIdx.x * blockDim.x + threadIdx.x;
  const int v = i & (DHW - 1);
  const int b = i >> 18;
  const int isroot = (L[i] == v + 1) ? 1 : 0;

  __shared__ int s[256];
  s[threadIdx.x] = isroot;
  __syncthreads();
  for (int o = 128; o > 0; o >>= 1) {
    if (threadIdx.x < o) s[threadIdx.x] += s[threadIdx.x + o];
    __syncthreads();
  }
  if (threadIdx.x == 0 && s[0] > 0) atomicAdd(&counts[b], s[0]);
}

// ---------------------------------------------------------------------------
// Kernel 6: per-batch point lookup + RMS. One 512-thread block per batch.
// ---------------------------------------------------------------------------
__global__ void rms_kernel(const float* __restrict__ pts,
                           const int* __restrict__ L,
                           const int* __restrict__ counts,
                           float* __restrict__ out) {
  const int b = blockIdx.x;
  const int i = threadIdx.x;             // 0..511

  __shared__ int   hits[PP];
  __shared__ float red[PP];

  const int pbase = (b * PP + i) * 3;
  const int d0 = (int)floorf(pts[pbase + 0]);
  const int d1 = (int)floorf(pts[pbase + 1]);
  const int d2 = (int)floorf(pts[pbase + 2]);
  const int lab = L[(b << 18) + (d0 << 12) + (d1 << 6) + d2];
  const int h = (lab > DHW) ? 0 : lab;
  hits[i] = h;
  __syncthreads();

  int cnt = 0;
  bool first = true;
  for (int j = 0; j < PP; ++j) {
    const int hj = hits[j];
    cnt += (hj == h);
    if (j < i && hj == h) first = false;
  }
  const bool pos = first && (h > 0);
  const float c0 = (h == 0) ? 1.0f : 0.0f;                       // -> hits0
  const float c1 = pos ? (1.0f - (float)cnt) * (1.0f - (float)cnt) : 0.0f;
  const float c2 = pos ? 1.0f : 0.0f;                            // -> n_hit_comp

  float hits0 = 0.f, sumsq = 0.f, nhit = 0.f;
  // three fixed-order tree reductions
  red[i] = c0; __syncthreads();
  for (int o = PP / 2; o > 0; o >>= 1) { if (i < o) red[i] += red[i + o]; __syncthreads(); }
  hits0 = red[0]; __syncthreads();

  red[i] = c1; __syncthreads();
  for (int o = PP / 2; o > 0; o >>= 1) { if (i < o) red[i] += red[i + o]; __syncthreads(); }
  sumsq = red[0]; __syncthreads();

  red[i] = c2; __syncthreads();
  for (int o = PP / 2; o > 0; o >>= 1) { if (i < o) red[i] += red[i + o]; __syncthreads(); }
  nhit = red[0];

  if (i == 0) {
    const float num = (float)counts[b];
    const float sq = hits0 * hits0 + sumsq + (num - nhit);
    out[b] = sqrtf(sq / (num + 1.0f));
  }
}

// ---------------------------------------------------------------------------
// Host launcher
// ---------------------------------------------------------------------------
extern "C" void kernel_launch(void* const* d_in, const int* in_sizes, int n_in,
                              void* d_out, int out_size, void* d_ws, size_t ws_size,
                              hipStream_t stream) {
  const float* logits = (const float*)d_in[0];   // (B,1,D,H,W) f32
  const float* pts    = (const float*)d_in[1];   // (B,P,3)     f32
  float* out = (float*)d_out;                    // (B,) f32

  const long n = (long)in_sizes[0];              // B*D*H*W

  // workspace layout
  int*   labels   = (int*)d_ws;                                  // 4 MiB
  float* partials = (float*)((char*)d_ws + (size_t)NTOT * 4);    // 64 f32
  float* meanp    = partials + NB_SUM;                           // 1 f32
  int*   counts   = (int*)(meanp + 1);                           // 4 i32

  // 1) global sum via WMMA matrix pipe
  sum_logits_wmma<<<NB_SUM, 256, 0, stream>>>(logits, n, partials);
  // 2) mean + zero counters
  finalize_mean<<<1, NB_SUM, 0, stream>>>(partials, meanp, counts, 1.0f / (float)n);
  // 3) init labels
  init_labels<<<NTOT / 256, 256, 0, stream>>>(logits, meanp, labels);
  // 4) fixed-count propagation (neighbor-min + pointer jumping)
  for (int it = 0; it < NPROP; ++it)
    propagate<<<NTOT / 256, 256, 0, stream>>>(labels);
  // 5) per-batch component counts
  count_roots<<<NTOT / 256, 256, 0, stream>>>(labels, counts);
  // 6) lookup + RMS
  rms_kernel<<<BB, PP, 0, stream>>>(pts, labels, counts, out);
}